// CustomNeighborSearch_6305011991077
// MI455X (gfx1250) — compile-verified
//
#include <hip/hip_runtime.h>

typedef __attribute__((ext_vector_type(2))) float v2f;
typedef __attribute__((ext_vector_type(8))) float v8f;

#define Q_PTS 16384
#define N_PTS 16384
#define NCHUNKS 16
#define WAVES_PER_BLOCK 8

// ---------------------------------------------------------------------------
// Kernel 1: zero the per-query float count accumulators (row_splits slot).
// ---------------------------------------------------------------------------
__global__ void __launch_bounds__(256)
nbs_zero_counts(float* __restrict__ counts)
{
    counts[blockIdx.x * 256 + threadIdx.x] = 0.0f;
}

// ---------------------------------------------------------------------------
// Kernel 2: ball query. One wave owns a 16-query row block and a 1024-wide
// N chunk; each iteration computes a 16x16 sqdist tile with one
// V_WMMA_F32_16X16X4_F32:  D = A(16x4) * B(4x16) + C(16x16)
//   A[m] = [-2qx, -2qy, -2qz, q^2]   B[:,n] = [dx, dy, dz, 1]   C[m][n] = d^2[n]
// Counts accumulate as floats (accf[r] += mask value), reduced across the
// 16 columns of each 16-lane half once per wave at the end.
// ---------------------------------------------------------------------------
__global__ void __launch_bounds__(256)
nbs_ball_query(const float* __restrict__ data,
               const float* __restrict__ queries,
               const float* __restrict__ radius_p,
               float* __restrict__ out)
{
    const int lane = threadIdx.x & 31;
    const int waveInBlock = threadIdx.x >> 5;
    const int gw = blockIdx.x * WAVES_PER_BLOCK + waveInBlock;

    const int qt = gw / NCHUNKS;          // query tile [0,1024)
    const int nc = gw % NCHUNKS;          // N chunk    [0,16)
    const int qbase = qt * 16;
    const int nbase0 = nc * (N_PTS / NCHUNKS);

    const float r  = radius_p[0];
    const float r2 = r * r;

    const int  m  = lane & 15;
    const bool hi = lane >= 16;

    // --- A operand: queries (constant across the N loop) -------------------
    const float qx = queries[3 * (qbase + m) + 0];
    const float qy = queries[3 * (qbase + m) + 1];
    const float qz = queries[3 * (qbase + m) + 2];
    const float q2 = qx * qx + qy * qy + qz * qz;

    v2f a;
    // lanes 0-15: VGPR0=K0, VGPR1=K1 ; lanes 16-31: VGPR0=K2, VGPR1=K3
    a.x = hi ? (-2.0f * qz) : (-2.0f * qx);
    a.y = hi ? q2 : (-2.0f * qy);

    float* __restrict__ maskOut = out;
    float* __restrict__ wOut = out + (size_t)Q_PTS * N_PTS + (size_t)(Q_PTS + 1);
    float* __restrict__ cntOut = out + (size_t)Q_PTS * N_PTS;

    float accf[8];
#pragma unroll
    for (int rr = 0; rr < 8; ++rr) accf[rr] = 0.0f;

    const int ntiles = (N_PTS / NCHUNKS) / 16;  // 64

    // software pipeline: preload tile 0's data point
    float dx = data[3 * (nbase0 + m) + 0];
    float dy = data[3 * (nbase0 + m) + 1];
    float dz = data[3 * (nbase0 + m) + 2];

    for (int t = 0; t < ntiles; ++t) {
        const int nbase = nbase0 + t * 16;
        const int n = nbase + m;

        const float d2 = dx * dx + dy * dy + dz * dz;
        v2f b;
        b.x = hi ? dz : dx;
        b.y = hi ? 1.0f : dy;

        v8f c;
#pragma unroll
        for (int rr = 0; rr < 8; ++rr) c[rr] = d2;

        // prefetch next tile's data point (overlaps WMMA + stores)
        {
            const int tn = (t + 1 < ntiles) ? (t + 1) : (ntiles - 1);
            const int nn = nbase0 + tn * 16 + m;
            dx = data[3 * nn + 0];
            dy = data[3 * nn + 1];
            dz = data[3 * nn + 2];
        }

        // D[m][n] = q2[m] + d2[n] - 2 q.d  == squared distance
        v8f sq = __builtin_amdgcn_wmma_f32_16x16x4_f32(
            false, a, false, b, (short)0, c, false, false);

#pragma unroll
        for (int rr = 0; rr < 8; ++rr) {
            const float s = sq[rr];
            const bool inr = (s <= r2);
            const float mv = inr ? 1.0f : 0.0f;
            const float wv = inr ? fmaxf(s, 0.0f) : 0.0f;

            accf[rr] += mv;  // one v_add_f32 per row

            const size_t row = (size_t)(qbase + rr + (hi ? 8 : 0));
            const size_t idx = row * (size_t)N_PTS + (size_t)n;
            // 2.1 GB streamed out once -> non-temporal hint
            __builtin_nontemporal_store(mv, maskOut + idx);
            __builtin_nontemporal_store(wv, wOut + idx);
        }
    }

    // reduce each accf[rr] across its 16-lane half (rows rr / rr+8)
#pragma unroll
    for (int rr = 0; rr < 8; ++rr) {
        accf[rr] += __shfl_xor(accf[rr], 1, 16);
        accf[rr] += __shfl_xor(accf[rr], 2, 16);
        accf[rr] += __shfl_xor(accf[rr], 4, 16);
        accf[rr] += __shfl_xor(accf[rr], 8, 16);
    }

    if (m == 0) {  // lanes 0 (low half rows) and 16 (high half rows)
#pragma unroll
        for (int rr = 0; rr < 8; ++rr) {
            atomicAdd(&cntOut[qbase + rr + (hi ? 8 : 0)], accf[rr]);
        }
    }
}

// ---------------------------------------------------------------------------
// Kernel 3: single-block scan of the 16384 float counts -> float row_splits
// (in place; each element is read into registers before its slot is
// overwritten).
// ---------------------------------------------------------------------------
__global__ void __launch_bounds__(256)
nbs_scan(float* __restrict__ out)
{
    float* rs = out + (size_t)Q_PTS * N_PTS;

    __shared__ int partial[256];
    const int t = threadIdx.x;
    const int seg = Q_PTS / 256;  // 64

    int local[64];
    int s = 0;
#pragma unroll 4
    for (int j = 0; j < seg; ++j) {
        local[j] = (int)rs[t * seg + j];   // counts are exact small ints
        s += local[j];
    }
    partial[t] = s;
    __syncthreads();

    // Hillis-Steele inclusive scan over 256 partials
    for (int off = 1; off < 256; off <<= 1) {
        int v = partial[t];
        int add = (t >= off) ? partial[t - off] : 0;
        __syncthreads();
        partial[t] = v + add;
        __syncthreads();
    }

    int excl = partial[t] - s;
    int total = partial[255];

    int run = excl;
    for (int j = 0; j < seg; ++j) {
        rs[t * seg + j] = (float)run;   // row_splits[i] = prefix before i
        run += local[j];
    }
    if (t == 255) rs[Q_PTS] = (float)total;
}

// ---------------------------------------------------------------------------
extern "C" void kernel_launch(void* const* d_in, const int* in_sizes, int n_in,
                              void* d_out, int out_size, void* d_ws, size_t ws_size,
                              hipStream_t stream)
{
    const float* data    = (const float*)d_in[0];  // [N,3]
    const float* queries = (const float*)d_in[1];  // [Q,3]
    const float* radius  = (const float*)d_in[2];  // scalar
    float* out = (float*)d_out;  // [Q*N mask][Q+1 row_splits][Q*N weights]

    float* counts = out + (size_t)Q_PTS * N_PTS;

    nbs_zero_counts<<<Q_PTS / 256, 256, 0, stream>>>(counts);

    const int totalWaves = (Q_PTS / 16) * NCHUNKS;            // 16384
    const int blocks = totalWaves / WAVES_PER_BLOCK;           // 2048
    nbs_ball_query<<<blocks, 256, 0, stream>>>(data, queries, radius, out);

    nbs_scan<<<1, 256, 0, stream>>>(out);
}